// GNNSAGEConv_3693671874806
// MI455X (gfx1250) — compile-verified
//
#include <hip/hip_runtime.h>

typedef __attribute__((ext_vector_type(2))) float v2f;
typedef __attribute__((ext_vector_type(8))) float v8f;

#define N_NODES 100000
#define D 128
#define SLOPE 0.2f

// ---------------------------------------------------------------------------
// Phase 1: edge scatter.  One wave (32 lanes) per edge; each lane moves one
// float4 (4 of the 128 features).  Coalesced 512B gather of x[src], coalesced
// f32 atomics into agg[dst] (agg lives in d_out).  Lane 0 bumps the degree.
// ---------------------------------------------------------------------------
__global__ void sage_scatter(const float* __restrict__ x,
                             const int*   __restrict__ ei,
                             float*       __restrict__ agg,
                             float*       __restrict__ cnt,
                             int E) {
    long long t = (long long)blockIdx.x * blockDim.x + threadIdx.x;
    int e    = (int)(t >> 5);
    int lane = (int)(t & 31);
    if (e >= E) return;
    int src = ei[e];                 // edge_index[0, e]
    int dst = ei[(long long)E + e];  // edge_index[1, e]

    float4 v = ((const float4*)(x + (long long)src * D))[lane];
    float* a = agg + (long long)dst * D + lane * 4;
    atomicAdd(a + 0, v.x);
    atomicAdd(a + 1, v.y);
    atomicAdd(a + 2, v.z);
    atomicAdd(a + 3, v.w);
    if (lane == 0) atomicAdd(cnt + dst, 1.0f);
}

// ---------------------------------------------------------------------------
// Phase 2: fused mean-normalize + dual GEMM + bias + LeakyReLU, on the f32
// WMMA path (V_WMMA_F32_16X16X4_F32).  One wave owns a 16-row tile and all
// 128 output columns (8 accumulators of 16 cols).  agg is read from d_out and
// the same rows are overwritten after the k-loops complete (race-free).
//
//   out[16x128] = (agg/deg)[16x128] @ Wl^T + x[16x128] @ Wr^T + bl
//
// A frag (16x4, MxK): lanes 0-15 -> K = kk+{0,1}; lanes 16-31 -> K = kk+{2,3}
// B frag (4x16, KxN): lane holds column n = o + (lane&15), same K split.
// C/D: VGPR j -> row j (lanes<16) / row 8+j (lanes>=16), col = lane&15 + o.
// ---------------------------------------------------------------------------
__global__ void __launch_bounds__(256)
sage_gemm(const float* __restrict__ x,
          const float* __restrict__ Wl,
          const float* __restrict__ bl,
          const float* __restrict__ Wr,
          const float* __restrict__ cnt,
          float* __restrict__ out /* agg on entry, result on exit */) {
    const int wave = (blockIdx.x * (blockDim.x >> 5)) + (threadIdx.x >> 5);
    const int lane = threadIdx.x & 31;
    const int r0   = wave * 16;
    if (r0 >= N_NODES) return;                 // wave-uniform: EXEC stays full

    const int lrow  = lane & 15;               // A row / B col within tile
    const int khalf = (lane >> 4) * 2;         // K sub-offset: 0 or 2
    const int row   = r0 + lrow;

    const float rc = 1.0f / fmaxf(cnt[row], 1.0f);   // mean; zero-degree -> 0

    v8f acc[8];
#pragma unroll
    for (int o = 0; o < 8; ++o) { v8f z = {}; acc[o] = z; }

    const float* aggRow = out + (long long)row * D;
    const float* xRow   = x   + (long long)row * D;

    // ---- agg/deg @ Wl^T ----
    for (int kk = 0; kk < D; kk += 4) {
        float2 t = *(const float2*)(aggRow + kk + khalf);
        v2f a; a.x = t.x * rc; a.y = t.y * rc;
#pragma unroll
        for (int o = 0; o < 8; ++o) {
            float2 w = *(const float2*)(Wl + (long long)(o * 16 + lrow) * D + kk + khalf);
            v2f b; b.x = w.x; b.y = w.y;
            acc[o] = __builtin_amdgcn_wmma_f32_16x16x4_f32(
                false, a, false, b, (short)0, acc[o], false, false);
        }
    }

    // ---- x @ Wr^T (accumulated into same tiles) ----
    for (int kk = 0; kk < D; kk += 4) {
        float2 t = *(const float2*)(xRow + kk + khalf);
        v2f a; a.x = t.x; a.y = t.y;
#pragma unroll
        for (int o = 0; o < 8; ++o) {
            float2 w = *(const float2*)(Wr + (long long)(o * 16 + lrow) * D + kk + khalf);
            v2f b; b.x = w.x; b.y = w.y;
            acc[o] = __builtin_amdgcn_wmma_f32_16x16x4_f32(
                false, a, false, b, (short)0, acc[o], false, false);
        }
    }

    // ---- epilogue: bias + LeakyReLU, overwrite agg rows with the result ----
#pragma unroll
    for (int o = 0; o < 8; ++o) {
        const int   col = o * 16 + lrow;
        const float bb  = bl[col];
#pragma unroll
        for (int j = 0; j < 8; ++j) {
            const int rr = r0 + j + (lane >> 4) * 8;
            float v = acc[o][j] + bb;
            v = (v > 0.0f) ? v : SLOPE * v;
            out[(long long)rr * D + col] = v;
        }
    }
}

// ---------------------------------------------------------------------------
extern "C" void kernel_launch(void* const* d_in, const int* in_sizes, int n_in,
                              void* d_out, int out_size, void* d_ws, size_t ws_size,
                              hipStream_t stream) {
    const float* x  = (const float*)d_in[0];
    const int*   ei = (const int*)  d_in[1];
    const float* Wl = (const float*)d_in[2];
    const float* bl = (const float*)d_in[3];
    const float* Wr = (const float*)d_in[4];
    float* out = (float*)d_out;       // doubles as the agg accumulator
    float* cnt = (float*)d_ws;        // 100000 floats (400 KB) of scratch
    const int E = in_sizes[1] / 2;

    // zero accumulator + counts (capturable memset nodes)
    hipMemsetAsync(d_out, 0, (size_t)N_NODES * D * sizeof(float), stream);
    hipMemsetAsync(d_ws,  0, (size_t)N_NODES * sizeof(float),     stream);

    // Phase 1: one wave per edge
    {
        long long tot = (long long)E * 32;
        int blocks = (int)((tot + 255) / 256);
        sage_scatter<<<blocks, 256, 0, stream>>>(x, ei, out, cnt, E);
    }

    // Phase 2: 6250 row tiles, 8 waves (tiles) per 256-thread block
    {
        int tiles  = (N_NODES + 15) / 16;
        int blocks = (tiles + 7) / 8;
        sage_gemm<<<blocks, 256, 0, stream>>>(x, Wl, bl, Wr, cnt, out);
    }
}